// PoseConsistencyLoss_81535659147746
// MI455X (gfx1250) — compile-verified
//
#include <hip/hip_runtime.h>
#include <hip/hip_bf16.h>
#include <stdint.h>

// ---------------------------------------------------------------------------
// PoseConsistencyLoss for MI455X (gfx1250).
// HBM-bound streaming reduction: 157 MB in, 24 B out (~6.7 us roofline).
// Data path: TDM tensor_load_to_lds bulk tiles (pred tile from wave 0,
// target tile from wave 1, concurrently), s_wait_tensorcnt, wave32 shuffle
// reduction, ds_add_f32 block atomics, global f32 atomics.
// TILE=64 -> ~78 KB LDS/block -> 4 workgroups resident per 320 KB WGP
// -> 4 independent TDM streams in flight per WGP to saturate 23.3 TB/s.
// ---------------------------------------------------------------------------

#define TILE    64       // poses per block, 1 thread per pose (2 waves)
#define PSTRIDE 150      // floats per pose (50 joints * 3)
#define NJ      50
#define NB      49       // bones with range constraints
#define NC      49       // supervised connections

typedef __attribute__((ext_vector_type(4))) unsigned int u32x4;
typedef __attribute__((ext_vector_type(8))) int          i32x8;
typedef __attribute__((ext_vector_type(4))) int          i32x4;

// Bone table: 9 base bones then 40 hand bones (dict insertion order; order
// irrelevant for the mean).
__constant__ int c_bone_s[NB] = {
  0,1,1,2,3,5,6,4,7,
  8,8,8,8,8,9,10,11,13,14,15,17,18,19,21,22,23,25,26,27,
  29,29,29,29,29,30,31,32,34,35,36,38,39,40,42,43,44,46,47,48};
__constant__ int c_bone_e[NB] = {
  1,2,5,3,4,6,7,8,29,
  9,13,17,21,25,10,11,12,14,15,16,18,19,20,22,23,24,26,27,28,
  30,34,38,42,46,31,32,33,35,36,37,39,40,41,43,44,45,47,48,49};
__constant__ float c_bone_min[NB] = {
  0.05f,0.10f,0.10f,0.20f,0.20f,0.20f,0.20f,0.05f,0.05f,
  0.01f,0.01f,0.01f,0.01f,0.01f,0.01f,0.01f,0.01f,0.01f,0.01f,
  0.01f,0.01f,0.01f,0.01f,0.01f,0.01f,0.01f,0.01f,0.01f,0.01f,
  0.01f,0.01f,0.01f,0.01f,0.01f,0.01f,0.01f,0.01f,0.01f,0.01f,
  0.01f,0.01f,0.01f,0.01f,0.01f,0.01f,0.01f,0.01f,0.01f,0.01f};
__constant__ float c_bone_max[NB] = {
  0.15f,0.20f,0.20f,0.35f,0.35f,0.35f,0.35f,0.15f,0.15f,
  0.08f,0.08f,0.08f,0.08f,0.08f,0.08f,0.08f,0.08f,0.08f,0.08f,
  0.08f,0.08f,0.08f,0.08f,0.08f,0.08f,0.08f,0.08f,0.08f,0.08f,
  0.08f,0.08f,0.08f,0.08f,0.08f,0.08f,0.08f,0.08f,0.08f,0.08f,
  0.08f,0.08f,0.08f,0.08f,0.08f,0.08f,0.08f,0.08f,0.08f,0.08f};

__constant__ int c_conn_s[NC] = {
  0,1,2,3,1,5,6,7,8,8,8,8,8,9,10,11,13,14,15,17,18,19,21,22,23,
  25,26,27,4,29,29,29,29,29,30,31,32,34,35,36,38,39,40,42,43,44,46,47,48};
__constant__ int c_conn_e[NC] = {
  1,2,3,4,5,6,7,8,9,13,17,21,25,10,11,12,14,15,16,18,19,20,22,23,24,
  26,27,28,29,30,34,38,42,46,31,32,33,35,36,37,39,40,41,43,44,45,47,48,49};

// ---------------------------------------------------------------------------
// TDM 2-D tile load: rows of PSTRIDE f32, row stride PSTRIDE, into LDS.
// tensor_rows < tile_rows => hardware zero-fills the OOB halo rows.
// Descriptor layout per CDNA5 ISA 8.3/8.4 (D# group0 128b, group1 256b).
// ---------------------------------------------------------------------------
__device__ __forceinline__ void tdm_load_tile(unsigned lds_byte_addr,
                                              const float* gsrc,
                                              unsigned tensor_rows,
                                              unsigned tile_rows)
{
  unsigned long long ga = (unsigned long long)(uintptr_t)gsrc;
  u32x4 g0;
  g0.x = 1u;                                              // count=1 (valid), user mode
  g0.y = lds_byte_addr;                                   // lds_addr
  g0.z = (unsigned)(ga & 0xFFFFFFFFull);                  // global_addr[31:0]
  g0.w = (unsigned)((ga >> 32) & 0x01FFFFFFull)           // global_addr[56:32]
       | (2u << 30);                                      // type = 2 ("image")
  i32x8 g1;
  g1[0] = (int)(2u << 16);                                // data_size=2 (4B); mask=0
  g1[1] = (int)(((unsigned)PSTRIDE & 0xFFFFu) << 16);     // tensor_dim0 lo16 (hi half)
  g1[2] = (int)((((unsigned)PSTRIDE >> 16) & 0xFFFFu)     // tensor_dim0 hi16
              | ((tensor_rows & 0xFFFFu) << 16));         // tensor_dim1 lo16
  g1[3] = (int)(((tensor_rows >> 16) & 0xFFFFu)           // tensor_dim1 hi16
              | (((unsigned)PSTRIDE & 0xFFFFu) << 16));   // tile_dim0 = 150
  g1[4] = (int)(tile_rows & 0xFFFFu);                     // tile_dim1; tile_dim2=0
  g1[5] = PSTRIDE;                                        // tensor_dim0_stride lo32
  g1[6] = 0;                                              // stride0 hi16 | stride1 lo16
  g1[7] = 0;                                              // stride1 hi32
  i32x4 gz = {0, 0, 0, 0};
#if __clang_major__ >= 23
  i32x8 gz8 = {0, 0, 0, 0, 0, 0, 0, 0};
  __builtin_amdgcn_tensor_load_to_lds(g0, g1, gz, gz, gz8, 0);
#else
  __builtin_amdgcn_tensor_load_to_lds(g0, g1, gz, gz, 0);
#endif
}

extern "C" __global__ void __launch_bounds__(TILE)
pose_loss_main(const float* __restrict__ pred,
               const float* __restrict__ targ,
               float* __restrict__ g_acc,
               int B, int S)
{
  extern __shared__ float smem[];        // [ (TILE+2)*150 pred | TILE*150 targ ]
  __shared__ float s_acc[8];

  const int t  = threadIdx.x;
  const int s0 = blockIdx.x * TILE;
  const int b  = blockIdx.y;

  if (t < 8) s_acc[t] = 0.f;

  // ---- bulk HBM -> LDS via Tensor Data Mover ----
  // Wave 0 issues the pred tile (with +2 halo, zero-fill clamped by
  // tensor_dim1 = S - s0); wave 1 issues the target tile concurrently.
  // Each wave waits on its own TENSORcnt, then the barrier publishes LDS.
  {
    const unsigned ldsBase = __builtin_amdgcn_groupstaticsize();
    if (t < 32) {
      const float* gp = pred + ((size_t)b * (size_t)S + (size_t)s0) * PSTRIDE;
      tdm_load_tile(ldsBase, gp, (unsigned)(S - s0), TILE + 2);
      __builtin_amdgcn_s_wait_tensorcnt(0);
    } else {
      const float* gt = targ + ((size_t)b * (size_t)S + (size_t)s0) * PSTRIDE;
      tdm_load_tile(ldsBase + (unsigned)((TILE + 2) * PSTRIDE * 4), gt,
                    TILE, TILE);
      __builtin_amdgcn_s_wait_tensorcnt(0);
    }
  }
  __syncthreads();

  const float* P = smem + (size_t)t * PSTRIDE;
  const float* T = smem + (size_t)(TILE + 2) * PSTRIDE + (size_t)t * PSTRIDE;
  const int s = s0 + t;

  float boneSum = 0.f, angSum = 0.f, symSum = 0.f;
  float velSum = 0.f, accSum = 0.f, supSum = 0.f;

  // ---- bone length range penalty ----
  for (int i = 0; i < NB; ++i) {
    const int a = c_bone_s[i] * 3, e = c_bone_e[i] * 3;
    const float dx = P[e] - P[a], dy = P[e+1] - P[a+1], dz = P[e+2] - P[a+2];
    const float l = sqrtf(dx*dx + dy*dy + dz*dz);
    boneSum += fmaxf(l - c_bone_max[i], 0.f) + fmaxf(c_bone_min[i] - l, 0.f);
  }

  // ---- joint angle penalty ----
  {
    const int ang_a[4] = {1, 1, 2, 5};
    const int ang_b[4] = {2, 5, 3, 6};
    const int ang_c[4] = {3, 6, 4, 7};
#pragma unroll
    for (int i = 0; i < 4; ++i) {
      const int A = ang_a[i]*3, Bj = ang_b[i]*3, C = ang_c[i]*3;
      const float v1x = P[A] - P[Bj], v1y = P[A+1] - P[Bj+1], v1z = P[A+2] - P[Bj+2];
      const float v2x = P[C] - P[Bj], v2y = P[C+1] - P[Bj+1], v2z = P[C+2] - P[Bj+2];
      const float n1 = sqrtf(v1x*v1x + v1y*v1y + v1z*v1z);
      const float n2 = sqrtf(v2x*v2x + v2y*v2y + v2z*v2z);
      const float inv = (1.f / fmaxf(n1, 1e-12f)) * (1.f / fmaxf(n2, 1e-12f));
      float c = (v1x*v2x + v1y*v2y + v1z*v2z) * inv;
      c = fminf(fmaxf(c, -1.f), 1.f);
      const float deg = acosf(c) * 57.29577951308232f;
      angSum += fmaxf(30.f - deg, 0.f) + fmaxf(deg - 150.f, 0.f);
    }
  }

  // ---- symmetry: L = [8..28, 2,3,4], R = [29..49, 5,6,7] ----
  for (int i = 0; i < 24; ++i) {
    const int l3 = ((i < 21) ? (8 + i)  : (i - 19)) * 3;   // 21->2, 22->3, 23->4
    const int r3 = ((i < 21) ? (29 + i) : (i - 16)) * 3;   // 21->5, 22->6, 23->7
    symSum += fabsf(P[l3] + P[r3])
            + 0.5f * (fabsf(P[l3+1] - P[r3+1]) + fabsf(P[l3+2] - P[r3+2]));
  }

  // ---- supervised: |pred bone len - target bone len| over connections ----
  for (int i = 0; i < NC; ++i) {
    const int a = c_conn_s[i] * 3, e = c_conn_e[i] * 3;
    const float dx = P[e] - P[a], dy = P[e+1] - P[a+1], dz = P[e+2] - P[a+2];
    const float pb = sqrtf(dx*dx + dy*dy + dz*dz);
    const float tx = T[e] - T[a], ty = T[e+1] - T[a+1], tz = T[e+2] - T[a+2];
    const float tb = sqrtf(tx*tx + ty*ty + tz*tz);
    supSum += fabsf(pb - tb);
  }

  // ---- temporal: vel at s (needs s+1), acc at s (needs s+1, s+2) ----
  if (s + 1 < S) {
    const bool hasAcc = (s + 2 < S);
    const float* P1 = P + PSTRIDE;
    const float* P2 = P + 2 * PSTRIDE;
    for (int j = 0; j < NJ * 3; j += 3) {
      const float ax = P[j],  ay = P[j+1],  az = P[j+2];
      const float bx = P1[j], by = P1[j+1], bz = P1[j+2];
      const float vx = bx - ax, vy = by - ay, vz = bz - az;
      velSum += sqrtf(vx*vx + vy*vy + vz*vz);
      if (hasAcc) {
        const float qx = (P2[j]   - bx) - vx;
        const float qy = (P2[j+1] - by) - vy;
        const float qz = (P2[j+2] - bz) - vz;
        accSum += sqrtf(qx*qx + qy*qy + qz*qz);
      }
    }
  }

  // ---- wave32 shuffle reduction, then LDS atomics, then global atomics ----
#pragma unroll
  for (int off = 16; off > 0; off >>= 1) {
    boneSum += __shfl_down(boneSum, off, 32);
    angSum  += __shfl_down(angSum,  off, 32);
    symSum  += __shfl_down(symSum,  off, 32);
    velSum  += __shfl_down(velSum,  off, 32);
    accSum  += __shfl_down(accSum,  off, 32);
    supSum  += __shfl_down(supSum,  off, 32);
  }
  if ((t & 31) == 0) {
    atomicAdd(&s_acc[0], boneSum);
    atomicAdd(&s_acc[1], angSum);
    atomicAdd(&s_acc[2], symSum);
    atomicAdd(&s_acc[3], velSum);
    atomicAdd(&s_acc[4], accSum);
    atomicAdd(&s_acc[5], supSum);
  }
  __syncthreads();
  if (t < 6) atomicAdd(&g_acc[t], s_acc[t]);
}

extern "C" __global__ void pose_loss_init(float* g_acc)
{
  if (threadIdx.x < 8) g_acc[threadIdx.x] = 0.f;
}

extern "C" __global__ void pose_loss_finalize(const float* __restrict__ g_acc,
                                              float* __restrict__ out,
                                              int B, int S)
{
  if (threadIdx.x == 0) {
    const float ps = (float)B * (float)S;
    const float bone = g_acc[0] / (ps * 49.f);
    const float ang  = g_acc[1] / (ps * 4.f);
    const float sym  = g_acc[2] / (ps * 24.f);
    const float temporal =
        g_acc[4] / ((float)B * (float)(S - 2) * 50.f) +
        0.5f * g_acc[3] / ((float)B * (float)(S - 1) * 50.f);
    const float sup = g_acc[5] / (ps * 49.f);
    out[0] = bone;
    out[1] = ang;
    out[2] = sym;
    out[3] = temporal;
    out[4] = sup;
    out[5] = bone + 0.5f * ang + 0.3f * sym + 0.2f * temporal + sup;
  }
}

extern "C" void kernel_launch(void* const* d_in, const int* in_sizes, int n_in,
                              void* d_out, int out_size, void* d_ws, size_t ws_size,
                              hipStream_t stream) {
  const float* pred = (const float*)d_in[0];
  const float* targ = (const float*)d_in[1];
  float* out   = (float*)d_out;
  float* g_acc = (float*)d_ws;

  const int B = 64;
  const int S = in_sizes[0] / (B * PSTRIDE);   // 2048

  pose_loss_init<<<1, 32, 0, stream>>>(g_acc);

  const size_t shbytes = (size_t)(TILE + 2 + TILE) * PSTRIDE * sizeof(float);
  dim3 grid(S / TILE, B);
  pose_loss_main<<<grid, TILE, shbytes, stream>>>(pred, targ, g_acc, B, S);

  pose_loss_finalize<<<1, 32, 0, stream>>>(g_acc, out, B, S);
}